// MLSTM_57887569216038
// MI455X (gfx1250) — compile-verified
//
#include <hip/hip_runtime.h>
#include <cstdint>
#include <cstddef>

#define DEVINL __device__ __forceinline__

typedef float v2f __attribute__((ext_vector_type(2)));
typedef float v8f __attribute__((ext_vector_type(8)));

// problem dims
static constexpr int T_DIM  = 1024;
static constexpr int B_DIM  = 64;
static constexpr int IN_DIM = 32;
static constexpr int EMB    = 64;
static constexpr int HID    = 128;
static constexpr int KF     = 16;
static constexpr int OUTD   = 32;
static constexpr int G4     = 512;   // 4*HID

// LDS padding (bank-conflict free for wave32 / 64 banks / b64 loads):
//  - A-row strides: stride % 8 == 4  -> lanes cover banks {4m,4m+1}, hi-half {4m+2,4m+3}
//  - pair-packed B strides (float2): stride % 32 == 16 -> half-waves hit disjoint 32-bank sets
static constexpr int HS     = 132;   // float stride for 128-wide A rows
static constexpr int XS     = 68;    // float stride for 64-wide A rows
static constexpr int IS     = 36;    // float stride for 32-wide A rows
static constexpr int BP512  = 528;   // float2 stride for N=512 pair-packed B
static constexpr int BP64   = 80;    // float2 stride for N=64  pair-packed B

DEVINL float sigm(float x) { return 1.0f / (1.0f + __expf(-x)); }

// D = A(16x4,f32) * B(4x16,f32) + C(16x16,f32), full wave32 op.
// A frag: lane ln holds row m=ln, k = k0 + 2*lh + {0,1}   (lh = lane>=16)
// B frag: lane ln holds col n=ln, k = k0 + 2*lh + {0,1}   (pair-packed float2)
// C/D   : VGPR r -> (M = r + 8*lh, N = ln)
DEVINL v8f wmma4(v2f a, v2f b, v8f c) {
  return __builtin_amdgcn_wmma_f32_16x16x4_f32(false, a, false, b, (short)0, c,
                                               false, false);
}

DEVINL v8f vzero8() {
  v8f z;
#pragma unroll
  for (int r = 0; r < 8; ++r) z[r] = 0.0f;
  return z;
}

__global__ void zero_kernel(float* __restrict__ p, int n) {
  int i = blockIdx.x * blockDim.x + threadIdx.x;
  if (i < n) p[i] = 0.0f;
}

// ---------------------------------------------------------------------------
// Kernel 1: fused  x = inputs@W_emb + b_emb ;  xg0 = x@Wx0 + b0
// one block per 64 rows of the flattened [T*B] row space
// ---------------------------------------------------------------------------
__global__ void __launch_bounds__(512)
embed_xg0_kernel(const float* __restrict__ in,   // [T*B][32]
                 const float* __restrict__ We,   // [32][64]
                 const float* __restrict__ be,   // [64]
                 const float* __restrict__ Wx0,  // [64][512]
                 const float* __restrict__ b0,   // [512]
                 float* __restrict__ xg)         // [T*B][512]
{
  extern __shared__ __align__(16) char smem[];
  float* inP = (float*)smem;                                   // [64][IS]
  v2f*   WeP = (v2f*)(smem + 64 * IS * 4);                     // [16][BP64]
  float* xb  = (float*)(smem + 64 * IS * 4 + 16 * BP64 * 8);   // [64][XS]
  v2f*   WxP = (v2f*)(smem + 64 * IS * 4 + 16 * BP64 * 8 + 64 * XS * 4); // [32][BP512]

  const int tid  = threadIdx.x;
  const int wave = tid >> 5;
  const int lane = tid & 31;
  const int lh   = lane >> 4;
  const int ln   = lane & 15;
  const int row0 = blockIdx.x * 64;

  // stage A rows
  for (int i = tid; i < 64 * 32; i += 512) {
    int r = i >> 5, c = i & 31;
    inP[r * IS + c] = in[(size_t)(row0 + r) * 32 + c];
  }
  // stage pair-packed W_emb
  for (int i = tid; i < 16 * 64; i += 512) {
    int kp = i >> 6, n = i & 63;
    v2f w; w.x = We[(2 * kp) * 64 + n]; w.y = We[(2 * kp + 1) * 64 + n];
    WeP[kp * BP64 + n] = w;
  }
  // stage pair-packed Wx0
  for (int i = tid; i < 32 * 512; i += 512) {
    int kp = i >> 9, n = i & 511;
    v2f w; w.x = Wx0[(2 * kp) * 512 + n]; w.y = Wx0[(2 * kp + 1) * 512 + n];
    WxP[kp * BP512 + n] = w;
  }
  __syncthreads();

  // GEMM1: x[64,64] = inP[64,32] @ We + be ; 16 tiles, one per wave
  {
    const int tm = wave >> 2, tn = wave & 3;
    const int col = tn * 16 + ln;
    const float bv = be[col];            // overlaps with k-loop, used in epilogue
    v8f acc = vzero8();
    const int am = tm * 16 + ln;
#pragma unroll
    for (int k0 = 0; k0 < 32; k0 += 4) {
      v2f a = *(const v2f*)&inP[am * IS + k0 + 2 * lh];
      v2f b = WeP[((k0 >> 1) + lh) * BP64 + col];
      acc = wmma4(a, b, acc);
    }
#pragma unroll
    for (int r = 0; r < 8; ++r)
      xb[(tm * 16 + r + 8 * lh) * XS + col] = acc[r] + bv;
  }
  __syncthreads();

  // GEMM2: xg[64,512] = xb[64,64] @ Wx0 + b0 ; 128 tiles, 8 per wave
  for (int task = wave; task < 128; task += 16) {
    const int tm = task >> 5, tn = task & 31;
    const int col = tn * 16 + ln;
    const float bv = b0[col];
    v8f acc = vzero8();
    const int am = tm * 16 + ln;
#pragma unroll
    for (int k0 = 0; k0 < 64; k0 += 4) {
      v2f a = *(const v2f*)&xb[am * XS + k0 + 2 * lh];
      v2f b = WxP[((k0 >> 1) + lh) * BP512 + col];
      acc = wmma4(a, b, acc);
    }
#pragma unroll
    for (int r = 0; r < 8; ++r)
      xg[(size_t)(row0 + tm * 16 + r + 8 * lh) * 512 + col] = acc[r] + bv;
  }
}

// ---------------------------------------------------------------------------
// Kernel 3: xg1 = x1[T*B,128] @ Wx1[128,512] + b1
// ---------------------------------------------------------------------------
__global__ void __launch_bounds__(512)
xg1_kernel(const float* __restrict__ x1,   // [T*B][128]
           const float* __restrict__ Wx1,  // [128][512]
           const float* __restrict__ b1,   // [512]
           float* __restrict__ xg)         // [T*B][512]
{
  extern __shared__ __align__(16) char smem[];
  float* aB = (float*)smem;                  // [64][HS]   33.8 KB
  v2f*   WP = (v2f*)(smem + 64 * HS * 4);    // [64][BP512] 264 KB

  const int tid  = threadIdx.x;
  const int wave = tid >> 5;
  const int lane = tid & 31;
  const int lh   = lane >> 4;
  const int ln   = lane & 15;
  const int row0 = blockIdx.x * 64;

  for (int i = tid; i < 64 * 128; i += 512) {
    int r = i >> 7, c = i & 127;
    aB[r * HS + c] = x1[(size_t)(row0 + r) * 128 + c];
  }
  for (int i = tid; i < 64 * 512; i += 512) {
    int kp = i >> 9, n = i & 511;
    v2f w; w.x = Wx1[(2 * kp) * 512 + n]; w.y = Wx1[(2 * kp + 1) * 512 + n];
    WP[kp * BP512 + n] = w;
  }
  __syncthreads();

  for (int task = wave; task < 128; task += 16) {
    const int tm = task >> 5, tn = task & 31;
    const int col = tn * 16 + ln;
    const float bv = b1[col];
    v8f acc = vzero8();
    const int am = tm * 16 + ln;
    for (int k0 = 0; k0 < 128; k0 += 4) {
      v2f a = *(const v2f*)&aB[am * HS + k0 + 2 * lh];
      v2f b = WP[((k0 >> 1) + lh) * BP512 + col];
      acc = wmma4(a, b, acc);
    }
#pragma unroll
    for (int r = 0; r < 8; ++r)
      xg[(size_t)(row0 + tm * 16 + r + 8 * lh) * 512 + col] = acc[r] + bv;
  }
}

// ---------------------------------------------------------------------------
// Kernels 2/4: sequential mLSTM scan. ONE persistent workgroup, 512 threads.
// LDS: Wh pair-packed (264 KB) + h state (33.8 KB)  < 320 KB/WGP.
// Ring buffer of K=16 past candidates lives in L2-resident global scratch.
// NOUT = 128 (layer 0 -> x1) or 32 (layer 1 -> final output)
// ---------------------------------------------------------------------------
template <int NOUT>
__global__ void __launch_bounds__(512)
scan_kernel(const float* __restrict__ xg,   // [T][64][512]  (bias included)
            const float* __restrict__ Wh,   // [128][512]
            const float* __restrict__ Wo,   // [128][NOUT]
            const float* __restrict__ bo,   // [NOUT]
            float* __restrict__ ring,       // [16][64][128], pre-zeroed
            float* __restrict__ out)        // [T][64][NOUT]
{
  extern __shared__ __align__(16) char smem[];
  v2f*   WhP  = (v2f*)smem;                      // [64][BP512] pair-packed, 264 KB
  float* hbuf = (float*)(smem + 64 * BP512 * 8); // [64][HS] h state, 33.8 KB

  const int tid  = threadIdx.x;
  const int wave = tid >> 5;
  const int lane = tid & 31;
  const int lh   = lane >> 4;
  const int ln   = lane & 15;

  // stage Wh pairs, zero h0
  for (int i = tid; i < 64 * 512; i += 512) {
    int kp = i >> 9, n = i & 511;
    v2f w; w.x = Wh[(2 * kp) * 512 + n]; w.y = Wh[(2 * kp + 1) * 512 + n];
    WhP[kp * BP512 + n] = w;
  }
  for (int i = tid; i < 64 * HS; i += 512) hbuf[i] = 0.0f;
  __syncthreads();

  float hnew[2][8];

  for (int t = 0; t < T_DIM; ++t) {
    const float* xgt = xg + (size_t)t * (64 * 512);

    // prefetch the next 4 timesteps' xg tiles (128 B lines) while we compute
    if (t + 1 < T_DIM) {
      const float* nxt = xgt + 64 * 512;
#pragma unroll
      for (int q = 0; q < 8; ++q)
        __builtin_prefetch(nxt + ((size_t)tid + q * 512) * 32, 0, 0);
    }

    // ---- phase 1: g = xg_t + h @ Wh ; gates ; fractional filter ; h_new ----
    // 32 tasks = (tm:4 x tnh:8); each wave owns 2; each task computes the
    // i/o/c/d gate tiles for the SAME (b,h) positions -> gates stay in regs.
#pragma unroll
    for (int ti = 0; ti < 2; ++ti) {
      const int task = wave + ti * 16;
      const int tm   = task >> 3;          // batch tile 0..3
      const int tnh  = task & 7;           // hidden tile 0..7
      const int colh = tnh * 16 + ln;      // h index

      // issue xg loads FIRST: their latency overlaps the whole WMMA k-loop
      // (zero-init accumulators; xg contribution added in the gate phase)
      float xv0[8], xv1[8], xv2[8], xv3[8];
#pragma unroll
      for (int r = 0; r < 8; ++r) {
        const int row = tm * 16 + r + 8 * lh;     // b index
        const float* p = xgt + row * 512 + colh;
        xv0[r] = p[0];  xv1[r] = p[128];
        xv2[r] = p[256]; xv3[r] = p[384];
      }

      v8f acc0 = vzero8(), acc1 = vzero8(), acc2 = vzero8(), acc3 = vzero8();
      const int am = tm * 16 + ln;
      for (int k0 = 0; k0 < 128; k0 += 4) {
        v2f a = *(const v2f*)&hbuf[am * HS + k0 + 2 * lh];   // reused 4x
        const int kp = (k0 >> 1) + lh;
        acc0 = wmma4(a, WhP[kp * BP512 + colh + 0],   acc0);
        acc1 = wmma4(a, WhP[kp * BP512 + colh + 128], acc1);
        acc2 = wmma4(a, WhP[kp * BP512 + colh + 256], acc2);
        acc3 = wmma4(a, WhP[kp * BP512 + colh + 384], acc3);
      }

      // gates (all in registers, lanes already aligned across i/o/c/d)
      float hc[8], og[8], dd[8];
#pragma unroll
      for (int r = 0; r < 8; ++r) {
        const float ig = sigm(acc0[r] + xv0[r]);
        og[r] = sigm(acc1[r] + xv1[r]);
        const float cg = tanhf(acc2[r] + xv2[r]);
        dd[r] = 0.5f * sigm(acc3[r] + xv3[r]);
        hc[r] = ig * cg;
      }

      // c_t = sum_{j=0}^{15} w_j(d) * hc_{t-j};  w_0=1, w_j = w_{j-1}*(j-1+d)/j
      float csum[8], w[8];
#pragma unroll
      for (int r = 0; r < 8; ++r) { csum[r] = hc[r]; w[r] = 1.0f; }
#pragma unroll
      for (int j = 1; j < KF; ++j) {
        const float* rp = ring + (size_t)((t - j) & 15) * (64 * 128);
        const float jm1 = (float)(j - 1), rj = 1.0f / (float)j;
#pragma unroll
        for (int r = 0; r < 8; ++r) {
          const int row = tm * 16 + r + 8 * lh;
          w[r] *= (jm1 + dd[r]) * rj;
          csum[r] += w[r] * rp[row * 128 + colh];
        }
      }
      // push newest candidate; same lane always owns same (b,h) -> no races,
      // and same-wave same-address store->load ordering covers t -> t+1 reads
      float* rw = ring + (size_t)(t & 15) * (64 * 128);
#pragma unroll
      for (int r = 0; r < 8; ++r) {
        const int row = tm * 16 + r + 8 * lh;
        rw[row * 128 + colh] = hc[r];
        hnew[ti][r] = og[r] * tanhf(csum[r]);
      }
    }

    __syncthreads();   // all waves done READING hbuf (old h)
#pragma unroll
    for (int ti = 0; ti < 2; ++ti) {
      const int task = wave + ti * 16;
      const int tm = task >> 3, tnh = task & 7;
      const int colh = tnh * 16 + ln;
#pragma unroll
      for (int r = 0; r < 8; ++r)
        hbuf[(tm * 16 + r + 8 * lh) * HS + colh] = hnew[ti][r];
    }
    __syncthreads();   // new h visible to everyone

    // ---- phase 2: out_t = h_new @ Wo + bo  (M=64, K=128, N=NOUT) ----
    float* ot = out + (size_t)t * (64 * NOUT);
    for (int task = wave; task < 4 * (NOUT / 16); task += 16) {
      const int tm  = task / (NOUT / 16);
      const int tno = task % (NOUT / 16);
      const int col = tno * 16 + ln;
      const float bv = bo[col];          // issued early, consumed in epilogue
      v8f acc = vzero8();
      const int am = tm * 16 + ln;
      for (int k0 = 0; k0 < 128; k0 += 4) {
        v2f a = *(const v2f*)&hbuf[am * HS + k0 + 2 * lh];
        v2f b;
        b.x = Wo[(k0 + 2 * lh) * NOUT + col];       // hot in WGP$/L2 (<=64 KB)
        b.y = Wo[(k0 + 2 * lh + 1) * NOUT + col];
        acc = wmma4(a, b, acc);
      }
#pragma unroll
      for (int r = 0; r < 8; ++r)
        ot[(tm * 16 + r + 8 * lh) * NOUT + col] = acc[r] + bv;
    }
    // no 3rd barrier needed: until the next iteration's first barrier, hbuf
    // is only ever read.
  }
}

// ---------------------------------------------------------------------------
extern "C" void kernel_launch(void* const* d_in, const int* in_sizes, int n_in,
                              void* d_out, int out_size, void* d_ws, size_t ws_size,
                              hipStream_t stream) {
  const float* inputs = (const float*)d_in[0];
  const float* W_emb  = (const float*)d_in[1];
  const float* b_emb  = (const float*)d_in[2];
  const float* Wx0    = (const float*)d_in[3];
  const float* Wh0    = (const float*)d_in[4];
  const float* b0     = (const float*)d_in[5];
  const float* Wo0    = (const float*)d_in[6];
  const float* bo0    = (const float*)d_in[7];
  const float* Wx1    = (const float*)d_in[8];
  const float* Wh1    = (const float*)d_in[9];
  const float* b1     = (const float*)d_in[10];
  const float* Wo1    = (const float*)d_in[11];
  const float* bo1    = (const float*)d_in[12];
  float* out = (float*)d_out;

  // workspace: xg [65536][512] (128 MB, reused for both layers, L2-resident)
  //            x1 [65536][128] (32 MB)  ring [16][64][128] (0.5 MB)
  float* xg   = (float*)d_ws;
  float* x1   = xg + (size_t)T_DIM * B_DIM * G4;
  float* ring = x1 + (size_t)T_DIM * B_DIM * HID;
  const int ringN = KF * B_DIM * HID;

  const size_t ldsE = (size_t)64 * IS * 4 + 16 * BP64 * 8 + 64 * XS * 4 + 32 * BP512 * 8;
  const size_t ldsX = (size_t)64 * HS * 4 + 64 * BP512 * 8;
  const size_t ldsS = (size_t)64 * BP512 * 8 + 64 * HS * 4;

  const int nblk = (T_DIM * B_DIM) / 64;   // 1024 row-blocks

  // embed + layer-0 input gates (parallel WMMA)
  embed_xg0_kernel<<<nblk, 512, ldsE, stream>>>(inputs, W_emb, b_emb, Wx0, b0, xg);
  // layer-0 scan (persistent single-WGP WMMA)
  zero_kernel<<<ringN / 256, 256, 0, stream>>>(ring, ringN);
  scan_kernel<HID><<<1, 512, ldsS, stream>>>(xg, Wh0, Wo0, bo0, ring, x1);
  // layer-1 input gates (parallel WMMA, reuses xg buffer)
  xg1_kernel<<<nblk, 512, ldsX, stream>>>(x1, Wx1, b1, xg);
  // layer-1 scan -> final output
  zero_kernel<<<ringN / 256, 256, 0, stream>>>(ring, ringN);
  scan_kernel<OUTD><<<1, 512, ldsS, stream>>>(xg, Wh1, Wo1, bo1, ring, out);
}